// Pool_SAGEConv_24618752541195
// MI455X (gfx1250) — compile-verified
//
#include <hip/hip_runtime.h>

// ---------------- problem constants ----------------
#define NN   50000
#define EE   800000
#define CIN  64
#define COUT 64
#define BN_EPS 1e-5f

// ---------------- tiling config ----------------
// Edge GEMM: M=E, K=64, N=64.  8 waves/block, 16-edge tile per wave.
#define WPB1   8
#define GRID1  500
#define NW1    (WPB1 * GRID1)     // 4000 waves
#define TILES1 (EE / 16)          // 50000 tiles (exact)
#define LDA1   68                 // 64 + 4 pad
#define LDB1   68

// Node GEMM: M=N, K=128, N=64.  2 waves/block (LDS budget <= 64KB).
#define WPB2   2
#define GRID2  512
#define NW2    (WPB2 * GRID2)     // 1024 waves
#define TILES2 (NN / 16)          // 3125 tiles (exact)
#define LDA2   132                // 128 + 4 pad
#define LDB2   132

// ---------------- workspace layout (float offsets) ----------------
#define AGG_OFF 0                               // N*64 floats (12.8 MB)
#define P1_OFF  (NN * 64)                       // NW1*128 partials
#define S1_OFF  (P1_OFF + NW1 * 128)            // 128 sums (sum|sumsq)
#define P2_OFF  (S1_OFF + 128)                  // NW2*128 partials
#define S2_OFF  (P2_OFF + NW2 * 128)            // 128 sums

typedef float v2f __attribute__((ext_vector_type(2)));
typedef float v8f __attribute__((ext_vector_type(8)));

// D = A(16xK) * B(Kx64) + bias, via V_WMMA_F32_16X16X4_F32.
// A staged row-major [16][LDA], B staged transposed [64 cols][LDB rows].
// Fragment layout (ISA 7.12.2): lane holds A row = lane%16, K pair = 2*(lane/16)+{0,1};
// B col = 16*t + lane%16, same K pair; C/D vgpr j -> row j+8*(lane/16), col 16*t+lane%16.
template <int LDA, int LDB, int KSTEPS>
__device__ __forceinline__ void tile_mma(const float* __restrict__ A,
                                         const float* __restrict__ B,
                                         int r, int half, const float bv[4],
                                         v8f acc[4]) {
#pragma unroll
  for (int t = 0; t < 4; ++t) {
    float b = bv[t];
    acc[t] = (v8f){b, b, b, b, b, b, b, b};
  }
#pragma unroll 8
  for (int k = 0; k < KSTEPS; ++k) {
    v2f a = *(const v2f*)&A[r * LDA + 4 * k + 2 * half];
#pragma unroll
    for (int t = 0; t < 4; ++t) {
      v2f b = *(const v2f*)&B[(t * 16 + r) * LDB + 4 * k + 2 * half];
      acc[t] = __builtin_amdgcn_wmma_f32_16x16x4_f32(
          false, a, false, b, (short)0, acc[t], false, false);
    }
  }
}

// ---------------- kernel 0: zero agg ----------------
__global__ void k_zero(float* __restrict__ p, int n) {
  int i = blockIdx.x * blockDim.x + threadIdx.x;
  if (i < n) p[i] = 0.0f;
}

// ---------------- edge GEMM body (shared by stats & apply passes) ----------------
__device__ __forceinline__ void stage_edge_tile(const float* __restrict__ x,
                                                const int* __restrict__ eidx,
                                                const float* __restrict__ ew,
                                                float coef, int e0, int r,
                                                int half, float* __restrict__ myA) {
  int e = e0 + r;                       // lanes r and r+16 stage row r, col halves
  int src = eidx[e];
  float s = 1.0f + coef * ew[e];
#pragma unroll
  for (int v = 0; v < 8; ++v) {
    float4 xv = *(const float4*)&x[src * 64 + half * 32 + v * 4];
    float4 sv = make_float4(xv.x * s, xv.y * s, xv.z * s, xv.w * s);
    *(float4*)&myA[r * LDA1 + half * 32 + v * 4] = sv;
  }
}

// ---------------- kernel 1: edge GEMM, accumulate BN sums ----------------
__global__ __launch_bounds__(WPB1 * 32) void k_edge_stats(
    const float* __restrict__ x, const int* __restrict__ eidx,
    const float* __restrict__ ew, const float* __restrict__ W,
    const float* __restrict__ bvec, const float* __restrict__ coefp,
    float* __restrict__ ws) {
  __shared__ float sW[64 * LDB1];
  __shared__ float sA[WPB1 * 16 * LDA1];
  int tid = threadIdx.x;
  for (int i = tid; i < 64 * 64; i += WPB1 * 32)
    sW[(i & 63) * LDB1 + (i >> 6)] = W[i];   // transpose pool_W into LDS
  __syncthreads();

  int wid = tid >> 5, lane = tid & 31, r = lane & 15, half = lane >> 4;
  float* myA = &sA[wid * 16 * LDA1];
  float coef = coefp[0];
  float bv[4];
#pragma unroll
  for (int t = 0; t < 4; ++t) bv[t] = bvec[t * 16 + r];

  float sum[4] = {0, 0, 0, 0}, sq[4] = {0, 0, 0, 0};
  int gw = blockIdx.x * WPB1 + wid;
  for (int tile = gw; tile < TILES1; tile += NW1) {
    stage_edge_tile(x, eidx, ew, coef, tile * 16, r, half, myA);
    v8f acc[4];
    tile_mma<LDA1, LDB1, 16>(myA, sW, r, half, bv, acc);
#pragma unroll
    for (int t = 0; t < 4; ++t)
#pragma unroll
      for (int j = 0; j < 8; ++j) {
        float v = acc[t][j];
        sum[t] += v;
        sq[t] += v * v;
      }
  }
  // combine the two half-wave row groups (deterministic, in-wave)
#pragma unroll
  for (int t = 0; t < 4; ++t) {
    sum[t] += __shfl_xor(sum[t], 16, 32);
    sq[t]  += __shfl_xor(sq[t], 16, 32);
  }
  if (half == 0) {
#pragma unroll
    for (int t = 0; t < 4; ++t) {
      ws[P1_OFF + gw * 128 + t * 16 + r]      = sum[t];
      ws[P1_OFF + gw * 128 + 64 + t * 16 + r] = sq[t];
    }
  }
}

// ---------------- kernel 2/5: deterministic fixed-order reduction ----------------
__global__ void k_reduce(const float* __restrict__ parts, float* __restrict__ out,
                         int nparts) {
  int c = threadIdx.x;  // 128 threads
  float a = 0.0f;
  for (int w = 0; w < nparts; ++w) a += parts[w * 128 + c];
  out[c] = a;
}

// ---------------- kernel 3: recompute edge GEMM, BN+relu, scatter-max ----------------
__global__ __launch_bounds__(WPB1 * 32) void k_edge_apply(
    const float* __restrict__ x, const int* __restrict__ eidx,
    const float* __restrict__ ew, const float* __restrict__ W,
    const float* __restrict__ bvec, const float* __restrict__ coefp,
    const float* __restrict__ gamma, const float* __restrict__ beta,
    float* __restrict__ ws) {
  __shared__ float sW[64 * LDB1];
  __shared__ float sA[WPB1 * 16 * LDA1];
  int tid = threadIdx.x;
  for (int i = tid; i < 64 * 64; i += WPB1 * 32)
    sW[(i & 63) * LDB1 + (i >> 6)] = W[i];
  __syncthreads();

  int wid = tid >> 5, lane = tid & 31, r = lane & 15, half = lane >> 4;
  float* myA = &sA[wid * 16 * LDA1];
  float coef = coefp[0];
  float bv[4], sc[4], sh[4];
#pragma unroll
  for (int t = 0; t < 4; ++t) {
    int ch = t * 16 + r;
    bv[t] = bvec[ch];
    float mean = ws[S1_OFF + ch] * (1.0f / EE);
    float var  = ws[S1_OFF + 64 + ch] * (1.0f / EE) - mean * mean;
    float rstd = rsqrtf(var + BN_EPS);
    sc[t] = gamma[ch] * rstd;
    sh[t] = beta[ch] - mean * sc[t];
  }
  unsigned int* aggU = (unsigned int*)(ws + AGG_OFF);

  int gw = blockIdx.x * WPB1 + wid;
  for (int tile = gw; tile < TILES1; tile += NW1) {
    int e0 = tile * 16;
    stage_edge_tile(x, eidx, ew, coef, e0, r, half, myA);
    v8f acc[4];
    tile_mma<LDA1, LDB1, 16>(myA, sW, r, half, bv, acc);
    int dj[8];
#pragma unroll
    for (int j = 0; j < 8; ++j) dj[j] = eidx[EE + e0 + j + 8 * half];
#pragma unroll
    for (int t = 0; t < 4; ++t) {
      int ch = t * 16 + r;
#pragma unroll
      for (int j = 0; j < 8; ++j) {
        float v = fmaxf(acc[t][j] * sc[t] + sh[t], 0.0f);
        // v >= 0: u32 bit-pattern compare == float compare; agg pre-zeroed.
        atomicMax(&aggU[(unsigned)dj[j] * 64u + (unsigned)ch], __float_as_uint(v));
      }
    }
  }
}

// ---------------- kernel 4: node GEMM [x||agg] @ final_W + b, BN sums ----------------
__global__ __launch_bounds__(WPB2 * 32) void k_final(
    const float* __restrict__ x, const float* __restrict__ agg,
    const float* __restrict__ W, const float* __restrict__ bvec,
    float* __restrict__ out, float* __restrict__ ws) {
  __shared__ float sW[64 * LDB2];          // final_W transposed: [64 cols][128 rows]
  __shared__ float sA[WPB2 * 16 * LDA2];   // 16 rows x 128 cols (x || agg)
  int tid = threadIdx.x;
  for (int i = tid; i < 128 * 64; i += WPB2 * 32)
    sW[(i & 63) * LDB2 + (i >> 6)] = W[i];
  __syncthreads();

  int wid = tid >> 5, lane = tid & 31, r = lane & 15, half = lane >> 4;
  float* myA = &sA[wid * 16 * LDA2];
  float bv[4];
#pragma unroll
  for (int t = 0; t < 4; ++t) bv[t] = bvec[t * 16 + r];

  float sum[4] = {0, 0, 0, 0}, sq[4] = {0, 0, 0, 0};
  int gw = blockIdx.x * WPB2 + wid;
  for (int tile = gw; tile < TILES2; tile += NW2) {
    int n0 = tile * 16;
    // lane stages row r; half 0 -> x cols [0,63], half 1 -> agg cols [64,127]
    const float* srcp = half ? agg : x;
#pragma unroll
    for (int v = 0; v < 16; ++v) {
      float4 t4 = *(const float4*)&srcp[(n0 + r) * 64 + v * 4];
      *(float4*)&myA[r * LDA2 + half * 64 + v * 4] = t4;
    }
    v8f acc[4];
    tile_mma<LDA2, LDB2, 32>(myA, sW, r, half, bv, acc);
#pragma unroll
    for (int t = 0; t < 4; ++t) {
      int ch = t * 16 + r;
#pragma unroll
      for (int j = 0; j < 8; ++j) {
        float v = acc[t][j];
        int row = n0 + j + 8 * half;
        out[row * 64 + ch] = v;             // pre-BN h
        sum[t] += v;
        sq[t] += v * v;
      }
    }
  }
#pragma unroll
  for (int t = 0; t < 4; ++t) {
    sum[t] += __shfl_xor(sum[t], 16, 32);
    sq[t]  += __shfl_xor(sq[t], 16, 32);
  }
  if (half == 0) {
#pragma unroll
    for (int t = 0; t < 4; ++t) {
      ws[P2_OFF + gw * 128 + t * 16 + r]      = sum[t];
      ws[P2_OFF + gw * 128 + 64 + t * 16 + r] = sq[t];
    }
  }
}

// ---------------- kernel 6: final BN + relu (in place on d_out) ----------------
__global__ void k_bn_relu(float* __restrict__ h, const float* __restrict__ sums,
                          const float* __restrict__ gamma,
                          const float* __restrict__ beta) {
  int i = blockIdx.x * blockDim.x + threadIdx.x;
  if (i >= NN * 64) return;
  int c = i & 63;
  float mean = sums[c] * (1.0f / NN);
  float var  = sums[64 + c] * (1.0f / NN) - mean * mean;
  float rstd = rsqrtf(var + BN_EPS);
  float scl = gamma[c] * rstd;
  float shf = beta[c] - mean * scl;
  h[i] = fmaxf(h[i] * scl + shf, 0.0f);
}

extern "C" void kernel_launch(void* const* d_in, const int* in_sizes, int n_in,
                              void* d_out, int out_size, void* d_ws, size_t ws_size,
                              hipStream_t stream) {
  (void)in_sizes; (void)n_in; (void)out_size; (void)ws_size;
  const float* x      = (const float*)d_in[0];
  const int*   eidx   = (const int*)d_in[1];
  const float* ew     = (const float*)d_in[2];
  const float* poolW  = (const float*)d_in[3];
  const float* poolB  = (const float*)d_in[4];
  const float* bnpG   = (const float*)d_in[5];
  const float* bnpB   = (const float*)d_in[6];
  const float* finW   = (const float*)d_in[7];
  const float* finB   = (const float*)d_in[8];
  const float* bnfG   = (const float*)d_in[9];
  const float* bnfB   = (const float*)d_in[10];
  const float* coef   = (const float*)d_in[11];
  float* ws  = (float*)d_ws;
  float* out = (float*)d_out;

  int zn = NN * 64;
  k_zero<<<(zn + 255) / 256, 256, 0, stream>>>(ws + AGG_OFF, zn);
  k_edge_stats<<<GRID1, WPB1 * 32, 0, stream>>>(x, eidx, ew, poolW, poolB, coef, ws);
  k_reduce<<<1, 128, 0, stream>>>(ws + P1_OFF, ws + S1_OFF, NW1);
  k_edge_apply<<<GRID1, WPB1 * 32, 0, stream>>>(x, eidx, ew, poolW, poolB, coef,
                                                bnpG, bnpB, ws);
  k_final<<<GRID2, WPB2 * 32, 0, stream>>>(x, ws + AGG_OFF, finW, finB, out, ws);
  k_reduce<<<1, 128, 0, stream>>>(ws + P2_OFF, ws + S2_OFF, NW2);
  k_bn_relu<<<(NN * 64 + 255) / 256, 256, 0, stream>>>(out, ws + S2_OFF, bnfG, bnfB);
}